// GraphThinkingGNN_38276748542534
// MI455X (gfx1250) — compile-verified
//
#include <hip/hip_runtime.h>

typedef __attribute__((ext_vector_type(2))) float v2f;
typedef __attribute__((ext_vector_type(8))) float v8f;
typedef __attribute__((ext_vector_type(4))) unsigned int u32x4;
typedef __attribute__((ext_vector_type(4))) int i32x4;
typedef __attribute__((ext_vector_type(8))) int i32x8;

#define HDIM 128
#define NG   64

__device__ __forceinline__ float relu1(float x) {
    // single v_med3_f32 instead of canonicalize+max pair
    return __builtin_amdgcn_fmed3f(x, 0.0f, __builtin_inff());
}

// ---------------------------------------------------------------------------
// utility kernels
// ---------------------------------------------------------------------------
__global__ void zero_f32(float* __restrict__ p, int n) {
    int i = blockIdx.x * blockDim.x + threadIdx.x;
    if (i < n) p[i] = 0.0f;
}

__global__ void count_deg(const int* __restrict__ dst, float* __restrict__ deg, int E) {
    int i = blockIdx.x * blockDim.x + threadIdx.x;
    if (i < E) atomicAdd(&deg[dst[i]], 1.0f);
}

// dis[] holds raw in-degree counts on entry; becomes deg^-0.5 (with self loop)
__global__ void finalize_deg(float* __restrict__ dis, float* __restrict__ selfn, int N) {
    int i = blockIdx.x * blockDim.x + threadIdx.x;
    if (i < N) {
        float d = dis[i] + 1.0f;
        float r = rsqrtf(d);
        dis[i]   = r;
        selfn[i] = r * r;
    }
}

__global__ void edge_norm(const int* __restrict__ src, const int* __restrict__ dst,
                          const float* __restrict__ dis, float* __restrict__ enorm, int E) {
    int i = blockIdx.x * blockDim.x + threadIdx.x;
    if (i < E) enorm[i] = dis[src[i]] * dis[dst[i]];
}

// ---------------------------------------------------------------------------
// Stage W[128,128] fp32 (64 KB) into LDS. Preferred path: one TDM
// tensor_load_to_lds issued by wave 0 (TENSORcnt-tracked DMA); fallback:
// cooperative vector loads.
// ---------------------------------------------------------------------------
__device__ __forceinline__ void stage_W_to_lds(float* ldsW, const float* __restrict__ W) {
#if defined(__gfx1250__) && __has_builtin(__builtin_amdgcn_tensor_load_to_lds) && \
    __has_builtin(__builtin_amdgcn_s_wait_tensorcnt)
    if (threadIdx.x < 32) {  // wave 0 only (TDM ignores EXEC; one issue per wave)
        const unsigned lds_off = (unsigned)(uintptr_t)(void*)ldsW;  // LDS byte offset
        const unsigned long long ga = (unsigned long long)(uintptr_t)W;
        // D# group 0: count=1 | lds_addr | global_addr[56:0] | type=2 ("image")
        u32x4 g0;
        g0.x = 1u;
        g0.y = lds_off;
        g0.z = (unsigned)(ga & 0xffffffffull);
        g0.w = (unsigned)((ga >> 32) & 0x1ffffffull) | (2u << 30);
        // D# group 1: data_size=4B; tensor 128x128; tile 128x128; dim0 stride 128
        i32x8 g1;
        g1[0] = (int)(2u << 16);    // wg_mask=0, data_size=2 (4 bytes), no pad/iter
        g1[1] = (int)(128u << 16);  // [47:32] atomic_bar=0, [63:48] tensor_dim0.lo=128
        g1[2] = (int)(128u << 16);  // [79:64] tensor_dim0.hi=0, [95:80] wait.. tensor_dim1.lo=128
        g1[3] = (int)(128u << 16);  // [111:96] tensor_dim1.hi=0, [127:112] tile_dim0=128
        g1[4] = 128;                // [143:128] tile_dim1=128, [159:144] tile_dim2=0
        g1[5] = 128;                // [191:160] tensor_dim0_stride.lo=128
        g1[6] = 0;                  // stride0.hi=0, tensor_dim1_stride.lo=0 (unused, 2D)
        g1[7] = 0;
        i32x4 gz = {0, 0, 0, 0};    // groups 2/3 unused for 2-D tile
#if __clang_major__ >= 23
        i32x8 gz8 = {0, 0, 0, 0, 0, 0, 0, 0};
        __builtin_amdgcn_tensor_load_to_lds(g0, g1, gz, gz, gz8, 0);
#else
        __builtin_amdgcn_tensor_load_to_lds(g0, g1, gz, gz, 0);
#endif
        __builtin_amdgcn_s_wait_tensorcnt(0);
    }
    __syncthreads();
#else
    for (int idx = threadIdx.x; idx < HDIM * HDIM; idx += 256)
        ldsW[idx] = W[idx];
    __syncthreads();
#endif
}

// ---------------------------------------------------------------------------
// Fused GCN GEMM + self-loop epilogue.
//   T[N,128]   = (RELU ? relu(A) : A) @ W[128,128]    (+= previous T if ACCUM)
//   Out[N,128] = T * selfn[:,None] + bias             (only if EPI)
// 256 threads = 8 wave32; wave w owns column tile w*16; block owns 4 row
// tiles against one LDS-staged W (64 KB). fp32 WMMA 16x16x4, K=128.
// ---------------------------------------------------------------------------
template <int RELU, int ACCUM, int EPI>
__launch_bounds__(256)
__global__ void gemm_wmma_f32(const float* __restrict__ A, const float* __restrict__ W,
                              float* __restrict__ T, float* __restrict__ Out,
                              const float* __restrict__ selfn,
                              const float* __restrict__ bias, int N) {
    __shared__ float ldsW[HDIM * HDIM];
    stage_W_to_lds(ldsW, W);

    const int lane = threadIdx.x & 31;
    const int wave = threadIdx.x >> 5;
    const int mr   = lane & 15;         // row in A-frag / col in B-frag
    const int kb   = (lane >> 4) << 1;  // lanes 0-15 -> K+0/1, lanes 16-31 -> K+2/3
    const int col  = wave * 16 + mr;
    const float bcol = EPI ? bias[col] : 0.0f;

#pragma unroll 1
    for (int mt = 0; mt < 4; ++mt) {
        const int tileM = (blockIdx.x * 4 + mt) * 16;
        if (tileM >= N) break;
        const int m = tileM + mr;
        const int mclamp = (m < N) ? m : (N - 1);   // clamp: loads always legal
        const float* arow = A + (size_t)mclamp * HDIM;

        v8f acc = {0.f, 0.f, 0.f, 0.f, 0.f, 0.f, 0.f, 0.f};
#pragma unroll
        for (int k0 = 0; k0 < HDIM; k0 += 4) {
            v2f a = *(const v2f*)(arow + k0 + kb);
            if (RELU) { a.x = relu1(a.x); a.y = relu1(a.y); }
            v2f b;
            b.x = ldsW[(k0 + kb)     * HDIM + col];
            b.y = ldsW[(k0 + kb + 1) * HDIM + col];
            acc = __builtin_amdgcn_wmma_f32_16x16x4_f32(false, a, false, b,
                                                        (short)0, acc, false, false);
        }

        // C/D layout: VGPR r -> M = r + (lane>=16 ? 8 : 0), N = lane&15
        const int row0 = tileM + ((lane >> 4) << 3);
#pragma unroll
        for (int r = 0; r < 8; ++r) {
            int row = row0 + r;
            if (row < N) {
                size_t o = (size_t)row * HDIM + col;
                float v = acc[r];
                if (ACCUM) v += T[o];
                T[o] = v;
                if (EPI) Out[o] = v * selfn[row] + bcol;
            }
        }
    }
}

// ---------------------------------------------------------------------------
// Edge scatter: grid-stride, one wave per edge iteration, lane owns 4 feats.
// Out[dst] += T[src] * enorm[e]; next iteration's source row is prefetched
// (global_prefetch_b8) to hide gather latency behind the atomics.
// ---------------------------------------------------------------------------
__launch_bounds__(256)
__global__ void scatter_edges(const float* __restrict__ T, float* __restrict__ Out,
                              const int* __restrict__ src, const int* __restrict__ dst,
                              const float* __restrict__ enorm, int E, int nwaves) {
    const int wid  = (int)(((size_t)blockIdx.x * blockDim.x + threadIdx.x) >> 5);
    const int lane = threadIdx.x & 31;
    for (int e = wid; e < E; e += nwaves) {
        const int en = e + nwaves;
        if (en < E) {
            int sn = src[en];
            __builtin_prefetch(T + (size_t)sn * HDIM + lane * 4, 0, 0);
        }
        const int   s = src[e];
        const int   d = dst[e];
        const float w = enorm[e];
        const float4 v = ((const float4*)(T + (size_t)s * HDIM))[lane];
        float* od = Out + (size_t)d * HDIM + lane * 4;
        atomicAdd(od + 0, v.x * w);
        atomicAdd(od + 1, v.y * w);
        atomicAdd(od + 2, v.z * w);
        atomicAdd(od + 3, v.w * w);
    }
}

// ---------------------------------------------------------------------------
// Mean-pool (with final ReLU). batch[] is sorted, so per-block LDS partials
// (64 graphs x 128 feats = 32 KB) then a handful of global atomics.
// ---------------------------------------------------------------------------
__launch_bounds__(128)
__global__ void pool_partial(const float* __restrict__ H, const int* __restrict__ batch,
                             float* __restrict__ sums, float* __restrict__ cnt,
                             int N, int chunk) {
    __shared__ float lsum[NG * HDIM];
    __shared__ float lcnt[NG];
    int tid = threadIdx.x;  // feature id
    for (int idx = tid; idx < NG * HDIM; idx += 128) lsum[idx] = 0.f;
    for (int g = tid; g < NG; g += 128) lcnt[g] = 0.f;
    __syncthreads();

    int start = blockIdx.x * chunk;
    int end   = min(start + chunk, N);
    for (int i = start; i < end; ++i) {
        int b = batch[i];
        float v = H[(size_t)i * HDIM + tid];
        lsum[b * HDIM + tid] += relu1(v);
        if (tid == 0) lcnt[b] += 1.0f;
    }
    __syncthreads();

    for (int idx = tid; idx < NG * HDIM; idx += 128) {
        float v = lsum[idx];
        if (v != 0.f) atomicAdd(&sums[idx], v);
    }
    for (int g = tid; g < NG; g += 128) {
        float c = lcnt[g];
        if (c != 0.f) atomicAdd(&cnt[g], c);
    }
}

__global__ void pool_div(const float* __restrict__ sums, const float* __restrict__ cnt,
                         float* __restrict__ out) {
    int i = blockIdx.x * blockDim.x + threadIdx.x;
    if (i < NG * HDIM) out[i] = sums[i] / fmaxf(cnt[i >> 7], 1.0f);
}

// ---------------------------------------------------------------------------
// host-side orchestration
// ---------------------------------------------------------------------------
extern "C" void kernel_launch(void* const* d_in, const int* in_sizes, int n_in,
                              void* d_out, int out_size, void* d_ws, size_t ws_size,
                              hipStream_t stream) {
    (void)n_in; (void)out_size; (void)ws_size;
    const float* x   = (const float*)d_in[0];
    const float* Wp  = (const float*)d_in[1];
    const float* bp  = (const float*)d_in[2];
    const float* Wr  = (const float*)d_in[3];
    const float* br  = (const float*)d_in[4];
    const float* Wo0 = (const float*)d_in[5];
    const float* bo0 = (const float*)d_in[6];
    const float* Wo1 = (const float*)d_in[7];
    const float* bo1 = (const float*)d_in[8];
    const int* esrc  = (const int*)d_in[9];
    const int* edst  = (const int*)d_in[10];
    const int* batch = (const int*)d_in[11];

    const int N = in_sizes[0] / HDIM;
    const int E = in_sizes[9];

    // workspace layout (floats)
    float* ws = (float*)d_ws;
    size_t off = 0;
    float* bufA  = ws + off; off += (size_t)N * HDIM;   // ping: layer outputs
    float* bufB  = ws + off; off += (size_t)N * HDIM;   // pong: GEMM result t
    float* enorm = ws + off; off += (size_t)E;
    float* dis   = ws + off; off += (size_t)N;          // deg counts -> deg^-0.5
    float* selfn = ws + off; off += (size_t)N;
    float* sums  = ws + off; off += NG * HDIM;
    float* cnt   = ws + off; off += NG;

    // --- normalization coefficients ---
    zero_f32    <<<(N + 255) / 256, 256, 0, stream>>>(dis, N);
    count_deg   <<<(E + 255) / 256, 256, 0, stream>>>(edst, dis, E);
    finalize_deg<<<(N + 255) / 256, 256, 0, stream>>>(dis, selfn, N);
    edge_norm   <<<(E + 255) / 256, 256, 0, stream>>>(esrc, edst, dis, enorm, E);

    const int mTiles   = (N + 15) / 16;
    const int gemmGrid = (mTiles + 3) / 4;
    const int nwaves   = (E + 3) / 4;                       // 4 edges per wave
    const int scGrid   = (nwaves * 32 + 255) / 256;

    // One GCN layer: fused GEMM(+self/bias epilogue into bufA) then scatter.
    auto gcn = [&](const float* in, const float* W, const float* b, int relu_in,
                   const float* in2, const float* W2) {
        if (in2) {
            // concat layer: pass 1 computes T only, pass 2 accumulates + epilogue
            gemm_wmma_f32<1, 0, 0><<<gemmGrid, 256, 0, stream>>>(
                in, W, bufB, nullptr, nullptr, nullptr, N);
            gemm_wmma_f32<1, 1, 1><<<gemmGrid, 256, 0, stream>>>(
                in2, W2, bufB, bufA, selfn, b, N);
        } else if (relu_in) {
            gemm_wmma_f32<1, 0, 1><<<gemmGrid, 256, 0, stream>>>(
                in, W, bufB, bufA, selfn, b, N);
        } else {
            gemm_wmma_f32<0, 0, 1><<<gemmGrid, 256, 0, stream>>>(
                in, W, bufB, bufA, selfn, b, N);
        }
        scatter_edges<<<scGrid, 256, 0, stream>>>(bufB, bufA, esrc, edst, enorm, E, nwaves);
    };

    // projection
    gcn(x, Wp, bp, /*relu_in=*/0, nullptr, nullptr);
    // 4 recurrent layers (input needs relu of previous output)
    for (int i = 0; i < 4; ++i)
        gcn(bufA, Wr + (size_t)i * HDIM * HDIM, br + i * HDIM, 1, nullptr, nullptr);
    // concat([h, relu(x)]) @ Wo0 == h@Wo0[0:128] + relu(x)@Wo0[128:256]
    gcn(bufA, Wo0, bo0, 1, x, Wo0 + HDIM * HDIM);
    // output layer
    gcn(bufA, Wo1, bo1, 1, nullptr, nullptr);

    // --- global mean pool (with final relu) ---
    zero_f32<<<(NG * HDIM + 255) / 256, 256, 0, stream>>>(sums, NG * HDIM);
    zero_f32<<<1, NG, 0, stream>>>(cnt, NG);
    const int chunk = 512;
    pool_partial<<<(N + chunk - 1) / chunk, 128, 0, stream>>>(bufA, batch, sums, cnt, N, chunk);
    pool_div<<<(NG * HDIM + 255) / 256, 256, 0, stream>>>(sums, cnt, (float*)d_out);
}